// GuidedAttentionLoss_51367808860403
// MI455X (gfx1250) — compile-verified
//
#include <hip/hip_runtime.h>
#include <stdint.h>

// Guided attention loss on MI455X (gfx1250, wave32).
//   out = (1/2^26) * sum_{b, t<T_b, k<N_b} (1 - exp(-((k - floor(N_b/T_b * t))/N_b)^2 / 0.32)) * A[b,k,t]
// A is [64, 512, 2048] f32, t contiguous -> one block per (b,k) row, lanes over t.
//
// Memory-bound: only live data is read (k<N_b rows, t<T_b chunks) -> ~151 MB avg,
// ~6.5us at 23.3 TB/s. Data path: CDNA5 async global->LDS B128 (ASYNCcnt) staging,
// guide computed on the fly (one v_exp_f32/elem), reductions via
// V_WMMA_F32_16X16X4_F32 ones-matrix trick + 4 ds_swizzle xor-adds.
// Two-pass deterministic reduction through d_ws (no float atomics).

#define B_SZ   64
#define N_MAXX 512
#define T_MAXX 2048

typedef float v2f __attribute__((ext_vector_type(2)));
typedef float v8f __attribute__((ext_vector_type(8)));

template <int OFFS>
__device__ __forceinline__ float swz_add(float s) {
    return s + __int_as_float(__builtin_amdgcn_ds_swizzle(__float_as_int(s), OFFS));
}

// Full-wave32 f32 sum: D = ones(16x4) x B(4x16) + 0 puts every column sum of B in
// each row of D; VGPR d[0] lanes 0..15 hold the 16 column sums (16..31 duplicate).
// Summing within each 16-lane group (xor 8,4,2,1) yields the exact total in all lanes.
__device__ __forceinline__ float wave_reduce_f32(float v) {
    v2f a; a.x = 1.0f; a.y = 1.0f;   // A = all ones
    v2f b; b.x = v;    b.y = 0.0f;   // one live partial per lane
    v8f c = {};
    c = __builtin_amdgcn_wmma_f32_16x16x4_f32(
            false, a, false, b, (short)0, c, false, false);
    float s = c[0];
    s = swz_add<0x201f>(s);  // xor 8  (group-of-32 swizzle, and=0x1f)
    s = swz_add<0x101f>(s);  // xor 4
    s = swz_add<0x081f>(s);  // xor 2
    s = swz_add<0x041f>(s);  // xor 1
    return s;
}

__global__ __launch_bounds__(256)
void ga_partial_kernel(const float* __restrict__ align,
                       const int*   __restrict__ in_len,
                       const int*   __restrict__ tgt_len,
                       float*       __restrict__ partials) {
    __shared__ __align__(16) float buf[T_MAXX];  // 8 KB row stage
    __shared__ float wsum[8];

    const int blk = blockIdx.x;
    const int b   = blk >> 9;            // / N_MAXX
    const int k   = blk & (N_MAXX - 1);
    const int tid = threadIdx.x;

    const int Nb = in_len[b];
    const int Tb = tgt_len[b];

    float acc = 0.0f;
    if (k < Nb) {   // block-uniform branch (EXEC full inside)
        const float* src = align + ((size_t)b * N_MAXX + k) * T_MAXX;
        const uint32_t lbase = (uint32_t)(uintptr_t)(&buf[0]);
        // Stage only the live prefix t < Tb: 16B per lane per chunk, per-lane gated.
#pragma unroll
        for (int c = 0; c < 2; ++c) {
            const int f0 = tid * 4 + c * 1024;          // first float this lane copies
            if (f0 < Tb) {
                const uint32_t off   = (uint32_t)(f0 * 4);
                const uint64_t gaddr = (uint64_t)(uintptr_t)src + off;
                const uint32_t laddr = lbase + off;
                asm volatile("global_load_async_to_lds_b128 %0, %1, off"
                             :: "v"(laddr), "v"(gaddr) : "memory");
            }
        }
        asm volatile("s_wait_asynccnt 0" ::: "memory");
        __syncthreads();

        const float fN    = (float)Nb;
        const float ratio = fN / (float)Tb;
        const float invN  = 1.0f / fN;
        const float fk    = (float)k;
        const float cexp  = -3.125f * 1.44269504088896340736f; // -log2(e)/(2*sigma^2)
        for (int t = tid; t < Tb; t += 256) {
            const float d = (fk - floorf(ratio * (float)t)) * invN;
            const float e = __builtin_amdgcn_exp2f(d * d * cexp);
            acc += buf[t] * (1.0f - e);
        }
    }

    // block reduce: wave WMMA-reduce -> LDS -> wave0 WMMA-reduce
    const float ws  = wave_reduce_f32(acc);
    const int wave = tid >> 5, lane = tid & 31;
    if (lane == 0) wsum[wave] = ws;
    __syncthreads();
    if (wave == 0) {   // uniform over wave 0
        const float v   = (lane < 8) ? wsum[lane] : 0.0f;  // select, no divergence
        const float tot = wave_reduce_f32(v);
        if (lane == 0) partials[blk] = tot;
    }
}

__global__ __launch_bounds__(1024)
void ga_final_kernel(const float* __restrict__ partials,
                     float*       __restrict__ out,
                     int n4, float scale) {
    __shared__ float wsum[32];
    const int tid = threadIdx.x;
    const float4* p4 = (const float4*)partials;
    float acc = 0.0f;
    for (int i = tid; i < n4; i += 1024) {
        const float4 v = p4[i];
        acc += (v.x + v.y) + (v.z + v.w);
    }
    const float ws  = wave_reduce_f32(acc);
    const int wave = tid >> 5, lane = tid & 31;
    if (lane == 0) wsum[wave] = ws;
    __syncthreads();
    if (wave == 0) {
        const float tot = wave_reduce_f32(wsum[lane]);
        if (lane == 0) out[0] = tot * scale;
    }
}

extern "C" void kernel_launch(void* const* d_in, const int* in_sizes, int n_in,
                              void* d_out, int out_size, void* d_ws, size_t ws_size,
                              hipStream_t stream) {
    (void)in_sizes; (void)n_in; (void)out_size; (void)ws_size;
    const float* align   = (const float*)d_in[0];
    const int*   in_len  = (const int*)d_in[1];
    const int*   tgt_len = (const int*)d_in[2];
    float*       parts   = (float*)d_ws;           // 32768 floats = 128 KB scratch

    const int nblocks = B_SZ * N_MAXX;             // 32768 rows of 2048 floats
    ga_partial_kernel<<<nblocks, 256, 0, stream>>>(align, in_len, tgt_len, parts);

    const float scale = 1.0f / 67108864.0f;        // 1 / (64*512*2048) = 2^-26 exact
    ga_final_kernel<<<1, 1024, 0, stream>>>(parts, (float*)d_out, nblocks / 4, scale);
}